// TAGCN_26714696581627
// MI455X (gfx1250) — compile-verified
//
#include <hip/hip_runtime.h>
#include <math.h>

typedef float v2f __attribute__((ext_vector_type(2)));
typedef float v8f __attribute__((ext_vector_type(8)));

// ---------------- elementwise / sparse kernels ----------------

__global__ void tag_zero_f32(float* __restrict__ p, long long n) {
    long long i = (long long)blockIdx.x * blockDim.x + threadIdx.x;
    if (i < n) p[i] = 0.0f;
}

__global__ void tag_degree(const int* __restrict__ col, float* __restrict__ deg, int E) {
    int e = blockIdx.x * blockDim.x + threadIdx.x;
    if (e < E) atomicAdd(&deg[col[e]], 1.0f);
}

// in place: deg -> dinv = deg>0 ? rsqrt(max(deg,1)) : 0
__global__ void tag_dinv(float* __restrict__ deg, int N) {
    int i = blockIdx.x * blockDim.x + threadIdx.x;
    if (i < N) {
        float d = deg[i];
        deg[i] = (d > 0.0f) ? rsqrtf(fmaxf(d, 1.0f)) : 0.0f;
    }
}

__global__ void tag_norm(const int* __restrict__ row, const int* __restrict__ col,
                         const float* __restrict__ dinv, float* __restrict__ nrm, int E) {
    int e = blockIdx.x * blockDim.x + threadIdx.x;
    if (e < E) nrm[e] = dinv[row[e]] * dinv[col[e]];
}

// h_out[col] += norm[e] * h_in[row], d=64 features.
// One work item = (edge, feature-quad): 16 quads of float4 per edge.
__global__ void tag_propagate64(const int* __restrict__ row, const int* __restrict__ col,
                                const float* __restrict__ nrm,
                                const float* __restrict__ hin, float* __restrict__ hout,
                                int E) {
    long long gid = (long long)blockIdx.x * blockDim.x + threadIdx.x;
    long long e = gid >> 4;
    int q = (int)(gid & 15);
    if (e >= E) return;
    int r = row[e];
    int c = col[e];
    float w = nrm[e];
    const float4 v = *reinterpret_cast<const float4*>(hin + (long long)r * 64 + q * 4);
    float* dst = hout + (long long)c * 64 + q * 4;
    atomicAdd(dst + 0, w * v.x);
    atomicAdd(dst + 1, w * v.y);
    atomicAdd(dst + 2, w * v.z);
    atomicAdd(dst + 3, w * v.w);
}

// ---------------- fused 4-hop WMMA GEMM ----------------
// OUT[r, c] = act( sum_{hop=0..3} (H[hop] @ W[hop])[r, c] + bias[c] )
// W layout: [4, DIN, DOUT] row-major. One wave per 16x16 output tile.
// V_WMMA_F32_16X16X4_F32: fp32 in, fp32 accumulate -> full precision.
//
// Branch-free K-loop: out-of-range rows (A) / cols (B) are CLAMPED, not
// masked. Row m of A only affects row m of D, col n of B only affects col n
// of D, so the garbage lands exclusively in D entries the store epilogue
// masks off. All loads are unconditional -> no exec-mask save/restore.
template <int DIN, int DOUT>
__global__ __launch_bounds__(32)
void tag_gemm4_wmma(const float* __restrict__ h0, const float* __restrict__ h1,
                    const float* __restrict__ h2, const float* __restrict__ h3,
                    const float* __restrict__ W, const float* __restrict__ bias,
                    float* __restrict__ OUT, int nrows, int do_relu) {
    const int row0 = blockIdx.x * 16;
    const int col0 = blockIdx.y * 16;
    const int lane = threadIdx.x;          // 0..31, full wave32 (EXEC all-1s)
    const int m    = lane & 15;            // A row / D col index within tile
    const int kb   = (lane >> 4) << 1;     // K sub-offset: lanes 0-15 -> 0, 16-31 -> 2

    int ar = row0 + m;  if (ar > nrows - 1) ar = nrows - 1;   // clamped A row
    const int cn = col0 + m;                                  // D/B column
    int cc = cn;        if (cc > DOUT - 1) cc = DOUT - 1;     // clamped B col

    const float* __restrict__ H[4] = {h0, h1, h2, h3};

    v8f acc = {};
    #pragma unroll
    for (int hop = 0; hop < 4; ++hop) {
        const float* __restrict__ ap = H[hop] + (size_t)ar * DIN + kb;        // A[ar][kb + k0..]
        const float* __restrict__ bp = W + (size_t)hop * DIN * DOUT
                                         + (size_t)kb * DOUT + cc;            // B[kb][cc]
        #pragma unroll
        for (int k0 = 0; k0 < DIN; k0 += 4) {
            v2f a = *reinterpret_cast<const v2f*>(ap + k0);  // b64 load, 8B aligned
            v2f b;
            b.x = bp[0];
            b.y = bp[DOUT];
            bp += 4 * DOUT;
            acc = __builtin_amdgcn_wmma_f32_16x16x4_f32(
                /*neg_a=*/false, a, /*neg_b=*/false, b,
                /*c_mod=*/(short)0, acc, /*reuse_a=*/false, /*reuse_b=*/false);
        }
    }

    // Epilogue: D VGPR v holds row (row0 + 8*(lane>>4) + v), col cn
    if (cn < DOUT) {
        const float bv = bias[cn];
        const int mbase = (lane >> 4) * 8;
        #pragma unroll
        for (int v = 0; v < 8; ++v) {
            int r = row0 + mbase + v;
            if (r < nrows) {
                float val = acc[v] + bv;
                if (do_relu) val = fmaxf(val, 0.0f);
                OUT[(size_t)r * DOUT + cn] = val;
            }
        }
    }
}

// ---------------- row-wise log_softmax, in place ----------------
__global__ void tag_log_softmax(float* __restrict__ out, int nrows, int dout) {
    int r = blockIdx.x * blockDim.x + threadIdx.x;
    if (r >= nrows) return;
    float* p = out + (long long)r * dout;
    float mx = -INFINITY;
    for (int j = 0; j < dout; ++j) mx = fmaxf(mx, p[j]);
    float s = 0.0f;
    for (int j = 0; j < dout; ++j) s += expf(p[j] - mx);
    float ls = logf(s);
    for (int j = 0; j < dout; ++j) p[j] = p[j] - mx - ls;
}

// ---------------- launch ----------------

extern "C" void kernel_launch(void* const* d_in, const int* in_sizes, int n_in,
                              void* d_out, int out_size, void* d_ws, size_t ws_size,
                              hipStream_t stream) {
    const float* x    = (const float*)d_in[0];   // [N,64]
    const int*   ei   = (const int*)  d_in[1];   // [2,E]
    const float* W1   = (const float*)d_in[2];   // [4,64,64]
    const float* b1   = (const float*)d_in[3];   // [64]
    const float* W2   = (const float*)d_in[4];   // [4,64,40]
    const float* b2   = (const float*)d_in[5];   // [40]
    float* out        = (float*)d_out;           // [N,40]

    const int d_in_f = 64, d_hid = 64, d_outf = 40;
    const int N = in_sizes[0] / d_in_f;
    const int E = in_sizes[1] / 2;
    const int* row = ei;        // edge_index[0]
    const int* col = ei + E;    // edge_index[1]

    // workspace carve-up (floats)
    float* ws   = (float*)d_ws;
    float* dinv = ws;                       // [N]   (deg, then dinv in place)
    float* nrm  = dinv + N;                 // [E]
    float* P1   = nrm + E;                  // [N,64]
    float* P2   = P1 + (size_t)N * 64;      // [N,64]
    float* P3   = P2 + (size_t)N * 64;      // [N,64]
    float* Y1   = P3 + (size_t)N * 64;      // [N,64]
    (void)ws_size; (void)n_in; (void)out_size;

    const int TB = 256;
    auto blocks = [](long long n, int tb) { return (unsigned)((n + tb - 1) / tb); };

    // --- normalization ---
    tag_zero_f32<<<blocks(N, TB), TB, 0, stream>>>(dinv, N);
    tag_degree<<<blocks(E, TB), TB, 0, stream>>>(col, dinv, E);
    tag_dinv<<<blocks(N, TB), TB, 0, stream>>>(dinv, N);
    tag_norm<<<blocks(E, TB), TB, 0, stream>>>(row, col, dinv, nrm, E);

    const long long propItems = (long long)E * 16;
    const long long featN = (long long)N * 64;

    // --- layer 1: hops ---
    tag_zero_f32<<<blocks(featN * 3, TB), TB, 0, stream>>>(P1, featN * 3); // P1,P2,P3 contiguous
    tag_propagate64<<<blocks(propItems, TB), TB, 0, stream>>>(row, col, nrm, x,  P1, E);
    tag_propagate64<<<blocks(propItems, TB), TB, 0, stream>>>(row, col, nrm, P1, P2, E);
    tag_propagate64<<<blocks(propItems, TB), TB, 0, stream>>>(row, col, nrm, P2, P3, E);
    // Y1 = relu( x@W1[0] + P1@W1[1] + P2@W1[2] + P3@W1[3] + b1 )
    {
        dim3 grid((N + 15) / 16, (d_hid + 15) / 16);
        tag_gemm4_wmma<64, 64><<<grid, 32, 0, stream>>>(x, P1, P2, P3, W1, b1, Y1,
                                                        N, /*relu=*/1);
    }

    // --- layer 2: hops ---
    tag_zero_f32<<<blocks(featN * 3, TB), TB, 0, stream>>>(P1, featN * 3);
    tag_propagate64<<<blocks(propItems, TB), TB, 0, stream>>>(row, col, nrm, Y1, P1, E);
    tag_propagate64<<<blocks(propItems, TB), TB, 0, stream>>>(row, col, nrm, P1, P2, E);
    tag_propagate64<<<blocks(propItems, TB), TB, 0, stream>>>(row, col, nrm, P2, P3, E);
    // out = Y1@W2[0] + P1@W2[1] + P2@W2[2] + P3@W2[3] + b2  (dout=40 -> 3 col tiles, guarded)
    {
        dim3 grid((N + 15) / 16, (d_outf + 15) / 16);
        tag_gemm4_wmma<64, 40><<<grid, 32, 0, stream>>>(Y1, P1, P2, P3, W2, b2, out,
                                                        N, /*relu=*/0);
    }

    // --- log_softmax in place on d_out ---
    tag_log_softmax<<<blocks(N, TB), TB, 0, stream>>>(out, N, d_outf);
}